// MeanModel_25469156065611
// MI455X (gfx1250) — compile-verified
//
#include <hip/hip_runtime.h>
#include <stdint.h>

// Problem constants (from reference): B=512, H=4096, T=16
#define TOF_B 512
#define TOF_H 4096
#define TOF_T 16

// ---------------- CDNA5 async global->LDS support ----------------
#if defined(__has_builtin)
#  if __has_builtin(__builtin_amdgcn_global_load_async_to_lds_b128)
#    define USE_ASYNC_LDS 1
#  endif
#endif
#ifndef USE_ASYNC_LDS
#  define USE_ASYNC_LDS 0
#endif

typedef int v4i __attribute__((ext_vector_type(4)));
typedef __attribute__((address_space(1))) v4i GlobalV4;
typedef __attribute__((address_space(3))) v4i LocalV4;

__device__ __forceinline__ void wait_async_zero() {
#if defined(__has_builtin) && __has_builtin(__builtin_amdgcn_s_wait_asynccnt)
    __builtin_amdgcn_s_wait_asynccnt(0);
#else
    asm volatile("s_wait_asynccnt 0x0" ::: "memory");
#endif
    // Compiler barrier: the async unit wrote LDS behind the compiler's back.
    asm volatile("" ::: "memory");
}

// ---------------- Phase 1: per-(b,t) mask + circular neighbor table ---------
// One block per image b. Deterministic LDS tree reduction of the 16 column
// sums over H=4096 rows, then each of the first 16 threads finds its nearest
// valid channel going +i (left) and -i (right) circularly, packing l|(r<<4).
__global__ __launch_bounds__(256)
void tof_mask_kernel(const float* __restrict__ x, int* __restrict__ desc) {
    const int b   = blockIdx.x;
    const int tid = threadIdx.x;

    __shared__ float    red[256 * 17];   // padded stride 17 dwords
    __shared__ unsigned vmask;

    float acc[16];
#pragma unroll
    for (int k = 0; k < 16; ++k) acc[k] = 0.0f;

    const float4* xb = (const float4*)(x + (size_t)b * (TOF_H * TOF_T));
#pragma unroll 4
    for (int it = 0; it < TOF_H / 256; ++it) {
        const int h = it * 256 + tid;
        const float4* rp = xb + h * 4;
        float4 a0 = rp[0], a1 = rp[1], a2 = rp[2], a3 = rp[3];
        acc[0]  += a0.x; acc[1]  += a0.y; acc[2]  += a0.z; acc[3]  += a0.w;
        acc[4]  += a1.x; acc[5]  += a1.y; acc[6]  += a1.z; acc[7]  += a1.w;
        acc[8]  += a2.x; acc[9]  += a2.y; acc[10] += a2.z; acc[11] += a2.w;
        acc[12] += a3.x; acc[13] += a3.y; acc[14] += a3.z; acc[15] += a3.w;
    }

#pragma unroll
    for (int k = 0; k < 16; ++k) red[tid * 17 + k] = acc[k];
    if (tid == 0) vmask = 0u;
    __syncthreads();

    // Deterministic tree reduction over the 256 partial vectors.
    for (int s = 128; s > 0; s >>= 1) {
        if (tid < s) {
#pragma unroll
            for (int k = 0; k < 16; ++k)
                red[tid * 17 + k] += red[(tid + s) * 17 + k];
        }
        __syncthreads();
    }

    // valid[t] <=> column sum != 0.0 (disabled columns are exactly zero).
    if (tid < 16 && red[tid] != 0.0f) atomicOr(&vmask, 1u << tid);
    __syncthreads();

    if (tid < 16) {
        const unsigned vm = vmask;
        int l = tid, r = tid;
        for (int i = 0; i < 16; ++i) {
            const int c = (tid + i) & 15;
            if ((vm >> c) & 1u) { l = c; break; }
        }
        for (int i = 0; i < 16; ++i) {
            const int c = (tid - i) & 15;
            if ((vm >> c) & 1u) { r = c; break; }
        }
        desc[b * 16 + tid] = l | (r << 4);
    }
}

// ---------------- Phase 2: streaming circular-mean fill ---------------------
// Block = 256 threads = 8 waves. Each wave owns 32 consecutive rows of one
// image: it async-stages its 32x16-float tile into LDS (rows padded to
// 20 dwords: 16B-aligned b128 LDS writes, <=2-way bank conflict on gathers),
// waits on ASYNCcnt, then every lane gathers x[l[t]], x[r[t]] from its LDS
// row and stores 0.5*(xl+xr) — exact identity for valid channels (l=r=t).
#define ROW_STRIDE 20                       // dwords per padded row
#define WAVE_LDS   (32 * ROW_STRIDE)        // dwords per wave tile

__global__ __launch_bounds__(256)
void tof_fill_kernel(const float* __restrict__ x,
                     const int* __restrict__ desc,
                     float* __restrict__ out) {
    __shared__ float tile[8 * WAVE_LDS];

    const int bb   = blockIdx.x;            // 0 .. B*16-1
    const int b    = bb >> 4;
    const int h0   = (bb & 15) << 8;        // 256 rows per block
    const int wave = threadIdx.x >> 5;
    const int lane = threadIdx.x & 31;

    const size_t rowBase = (size_t)b * TOF_H + h0 + wave * 32;
    const float* gp = x + rowBase * TOF_T;  // wave tile: 32 rows * 64B = 2KB
    float* tl = &tile[wave * WAVE_LDS];

#if USE_ASYNC_LDS
#pragma unroll
    for (int j = 0; j < 4; ++j) {
        const int e   = lane * 4 + j * 128;     // element in 512-elem tile
        const int row = e >> 4, col = e & 15;   // col in {0,4,8,12}
        __builtin_amdgcn_global_load_async_to_lds_b128(
            (GlobalV4*)(float*)(gp + e),
            (LocalV4*)(tl + row * ROW_STRIDE + col),
            0, 0);
    }
    wait_async_zero();
#else
#pragma unroll
    for (int j = 0; j < 4; ++j) {
        const int e   = lane * 4 + j * 128;
        const int row = e >> 4, col = e & 15;
        const float4 v = *(const float4*)(gp + e);
        *(float4*)(tl + row * ROW_STRIDE + col) = v;   // 16B-aligned
    }
#endif

    const int*   db = desc + b * 16;        // wave-uniform -> scalar loads
    const float* rb = tl + lane * ROW_STRIDE;

    float v[16];
#pragma unroll
    for (int t = 0; t < 16; ++t) {
        const int d = db[t];
        const int l = d & 15;
        const int r = (d >> 4) & 15;
        v[t] = 0.5f * (rb[l] + rb[r]);
    }

    float4* o4 = (float4*)(out + (rowBase + lane) * TOF_T);
    o4[0] = make_float4(v[0],  v[1],  v[2],  v[3]);
    o4[1] = make_float4(v[4],  v[5],  v[6],  v[7]);
    o4[2] = make_float4(v[8],  v[9],  v[10], v[11]);
    o4[3] = make_float4(v[12], v[13], v[14], v[15]);
}

// ---------------- Host launcher --------------------------------------------
extern "C" void kernel_launch(void* const* d_in, const int* in_sizes, int n_in,
                              void* d_out, int out_size, void* d_ws, size_t ws_size,
                              hipStream_t stream) {
    const float* x    = (const float*)d_in[0];   // (B, H*T) f32
    // d_in[1] = tof_count (==16), baked in at compile time.
    int*   desc = (int*)d_ws;                    // B*T ints = 32 KB scratch
    float* out  = (float*)d_out;

    tof_mask_kernel<<<TOF_B, 256, 0, stream>>>(x, desc);
    tof_fill_kernel<<<TOF_B * (TOF_H / 256), 256, 0, stream>>>(x, desc, out);
}